// TaskAlignedAssigner_76158360093000
// MI455X (gfx1250) — compile-verified
//
#include <hip/hip_runtime.h>
#include <math.h>

#define BB   4
#define LL   8400
#define NG   16
#define NCLS 80
#define TOPKK 13

typedef __attribute__((ext_vector_type(2))) float v2f;
typedef __attribute__((ext_vector_type(8))) float v8f;

// ---------------------------------------------------------------------------
// K1: bbox_scores = one_hot(labels) @ pd_scores^T  via V_WMMA_F32_16X16X4_F32
// One wave (32 lanes) per (batch, 16-anchor tile). Exact in f32.
// A (16x4): lane = M (0..15), half = K-pair select. B (4x16): lane = N.
// ---------------------------------------------------------------------------
__global__ void k_bbox_scores_wmma(const float* __restrict__ pd_scores,
                                   const int* __restrict__ gt_labels,
                                   float* __restrict__ bbox_scores) {
    const int blk  = blockIdx.x;
    const int b    = blk / (LL / 16);
    const int l0   = (blk % (LL / 16)) * 16;
    const int lane = threadIdx.x;      // 0..31, EXEC all ones
    const int half = lane >> 4;        // selects K pair {0,1} vs {2,3}
    const int lo   = lane & 15;        // M for A, N for B

    const int lab = gt_labels[b * NG + lo];
    const float* bp = pd_scores + ((size_t)b * LL + (size_t)(l0 + lo)) * NCLS + 2 * half;

    v8f acc = {};
#pragma unroll
    for (int k0 = 0; k0 < NCLS; k0 += 4) {
        v2f a;
        a.x = (lab == k0 + 2 * half)     ? 1.0f : 0.0f;
        a.y = (lab == k0 + 2 * half + 1) ? 1.0f : 0.0f;
        v2f bv = *(const v2f*)(bp + k0);   // 8B contiguous, aligned
        acc = __builtin_amdgcn_wmma_f32_16x16x4_f32(
            false, a, false, bv, (short)0, acc, false, false);
    }
    // D layout: lanes 0-15: M=r, N=lane; lanes 16-31: M=r+8, N=lane-16
#pragma unroll
    for (int r = 0; r < 8; ++r) {
        int m = r + 8 * half;
        bbox_scores[((size_t)b * NG + m) * LL + l0 + lo] = acc[r];
    }
}

// ---------------------------------------------------------------------------
// K2: per (b,g,l): rotated IoU (polygon clip), mask_in, align.
// All local arrays accessed with compile-time indices -> stays in VGPRs.
// ---------------------------------------------------------------------------
__device__ __forceinline__ float xprod(float ax, float ay, float bx, float by) {
    return ax * by - ay * bx;
}

__global__ void k_pairs(const float* __restrict__ pd_bboxes,
                        const float* __restrict__ anc,
                        const float* __restrict__ gt_bboxes,
                        const float* __restrict__ bbox_scores,
                        float* __restrict__ overlaps,
                        float* __restrict__ align_raw,
                        float* __restrict__ alignm,
                        float* __restrict__ mask_in,
                        float* __restrict__ mask_pos) {
    const long long idx = (long long)blockIdx.x * 256 + threadIdx.x;
    if (idx >= (long long)BB * NG * LL) return;
    const int l = (int)(idx % LL);
    const int g = (int)((idx / LL) % NG);
    const int b = (int)(idx / ((long long)NG * LL));

    const float* gb = gt_bboxes + ((size_t)b * NG + g) * 5;
    const float* pb = pd_bboxes + ((size_t)b * LL + l) * 5;
    const float gcx = gb[0], gcy = gb[1], gw = gb[2], gh = gb[3];
    const float pcx = pb[0], pcy = pb[1], pw = pb[2], ph = pb[3];
    const float gc = cosf(gb[4]), gs = sinf(gb[4]);
    const float pc = cosf(pb[4]), ps = sinf(pb[4]);

    float ax[4], ay[4], bx[4], by[4];
    const float sx[4] = {0.5f, -0.5f, -0.5f, 0.5f};
    const float sy[4] = {0.5f, 0.5f, -0.5f, -0.5f};
#pragma unroll
    for (int i = 0; i < 4; ++i) {
        float dx = sx[i] * gw, dy = sy[i] * gh;
        ax[i] = gcx + gc * dx - gs * dy;
        ay[i] = gcy + gs * dx + gc * dy;
        float ex = sx[i] * pw, ey = sy[i] * ph;
        bx[i] = pcx + pc * ex - ps * ey;
        by[i] = pcy + ps * ex + pc * ey;
    }

    float px_[24], py_[24], mm[24];
#pragma unroll
    for (int i = 0; i < 4; ++i) {   // corners of A inside B (eps 1e-6, <=)
        float rx = ax[i] - pcx, ry = ay[i] - pcy;
        float qx = pc * rx + ps * ry, qy = -ps * rx + pc * ry;
        mm[i] = (fabsf(qx) <= pw * 0.5f + 1e-6f && fabsf(qy) <= ph * 0.5f + 1e-6f) ? 1.f : 0.f;
        px_[i] = ax[i]; py_[i] = ay[i];
    }
#pragma unroll
    for (int i = 0; i < 4; ++i) {   // corners of B inside A
        float rx = bx[i] - gcx, ry = by[i] - gcy;
        float qx = gc * rx + gs * ry, qy = -gs * rx + gc * ry;
        mm[4 + i] = (fabsf(qx) <= gw * 0.5f + 1e-6f && fabsf(qy) <= gh * 0.5f + 1e-6f) ? 1.f : 0.f;
        px_[4 + i] = bx[i]; py_[4 + i] = by[i];
    }
#pragma unroll
    for (int ia = 0; ia < 4; ++ia) {   // 16 edge-edge intersections
        int ia2 = (ia + 1) & 3;
        float d1x = ax[ia2] - ax[ia], d1y = ay[ia2] - ay[ia];
#pragma unroll
        for (int ib = 0; ib < 4; ++ib) {
            int ib2 = (ib + 1) & 3;
            float d2x = bx[ib2] - bx[ib], d2y = by[ib2] - by[ib];
            float den  = xprod(d1x, d1y, d2x, d2y);
            float dens = (fabsf(den) > 1e-9f) ? den : 1.0f;
            float dfx = bx[ib] - ax[ia], dfy = by[ib] - ay[ia];
            float t = xprod(dfx, dfy, d2x, d2y) / dens;
            float u = xprod(dfx, dfy, d1x, d1y) / dens;
            bool ok = (fabsf(den) > 1e-9f) && t >= 0.f && t <= 1.f && u >= 0.f && u <= 1.f;
            int j = 8 + ia * 4 + ib;
            px_[j] = ax[ia] + t * d1x;
            py_[j] = ay[ia] + t * d1y;
            mm[j]  = ok ? 1.f : 0.f;
        }
    }

    float nv = 0.f, cxs = 0.f, cys = 0.f;
#pragma unroll
    for (int i = 0; i < 24; ++i) { nv += mm[i]; cxs += px_[i] * mm[i]; cys += py_[i] * mm[i]; }
    float inv = 1.0f / fmaxf(nv, 1.0f);
    float ccx = cxs * inv, ccy = cys * inv;

    float ang[24];
#pragma unroll
    for (int i = 0; i < 24; ++i)
        ang[i] = (mm[i] > 0.f) ? atan2f(py_[i] - ccy, px_[i] - ccx) : 1e6f;

    // Stable ascending order via repeated min-extraction (all indices static,
    // no scratch). Shoelace accumulated on the fly: sum_{k>=1} cross(q_{k-1}-q_0, q_k-q_0)*m_k
    unsigned used = 0u;
    float p0x = 0.f, p0y = 0.f, pvx = 0.f, pvy = 0.f, ssum = 0.f;
    for (int k = 0; k < 24; ++k) {
        float bestA = 1e30f, bxp = 0.f, byp = 0.f, bmv = 0.f;
        int bi = 0;
#pragma unroll
        for (int i = 0; i < 24; ++i) {
            bool free_ = !((used >> i) & 1u);
            if (free_ && ang[i] < bestA) { bestA = ang[i]; bi = i; bxp = px_[i]; byp = py_[i]; bmv = mm[i]; }
        }
        used |= (1u << bi);
        if (k == 0) { p0x = bxp; p0y = byp; pvx = bxp; pvy = byp; }
        else {
            float v1x = pvx - p0x, v1y = pvy - p0y;
            float v2x = bxp - p0x, v2y = byp - p0y;
            ssum += xprod(v1x, v1y, v2x, v2y) * bmv;
            pvx = bxp; pvy = byp;
        }
    }
    float inter = (nv >= 3.0f) ? 0.5f * fabsf(ssum) : 0.0f;
    float iou = inter / (gw * gh + pw * ph - inter + 1e-9f);
    iou = fminf(fmaxf(iou, 0.0f), 1.0f);

    // mask_in: anchor center strictly inside GT (no eps)
    float arx = anc[l * 2 + 0] - gcx, ary = anc[l * 2 + 1] - gcy;
    float aqx = gc * arx + gs * ary, aqy = -gs * arx + gc * ary;
    float min_ = (fabsf(aqx) < gw * 0.5f && fabsf(aqy) < gh * 0.5f) ? 1.f : 0.f;

    float bs = bbox_scores[idx];
    float i2 = iou * iou;
    float al = bs * (i2 * i2 * i2);      // ALPHA=1, BETA=6

    overlaps[idx]  = iou;
    align_raw[idx] = al;
    alignm[idx]    = al * min_;
    mask_in[idx]   = min_;
    mask_pos[idx]  = 0.0f;               // pre-zero for K3 scatter
}

// ---------------------------------------------------------------------------
// K3: per (b,g) row: top-13 of align*mask_in (JAX tie-break: smaller index),
// scatter mask_pos = mask_in at picked anchors when mask_gt>0.
// Row staged in LDS (33.6KB of the 320KB WGP pool).
// ---------------------------------------------------------------------------
__global__ void k_topk(const float* __restrict__ alignm,
                       const float* __restrict__ mask_in,
                       const float* __restrict__ mask_gt,
                       float* __restrict__ mask_pos) {
    const int row = blockIdx.x;          // b*NG+g
    __shared__ float sv[LL];
    __shared__ float rv[256];
    __shared__ int   ri[256];
    __shared__ int   picks[TOPKK];
    const int tid = threadIdx.x;

    const float* src = alignm + (size_t)row * LL;
    for (int i = tid; i < LL; i += 256) sv[i] = src[i];
    __syncthreads();

    for (int k = 0; k < TOPKK; ++k) {
        float bv = -1e30f; int bi = 0;
        for (int i = tid; i < LL; i += 256) {
            float v = sv[i];
            if (v > bv) { bv = v; bi = i; }      // strict > keeps smallest i
        }
        rv[tid] = bv; ri[tid] = bi;
        __syncthreads();
        for (int s = 128; s > 0; s >>= 1) {
            if (tid < s) {
                float v2 = rv[tid + s]; int i2 = ri[tid + s];
                if (v2 > rv[tid] || (v2 == rv[tid] && i2 < ri[tid])) { rv[tid] = v2; ri[tid] = i2; }
            }
            __syncthreads();
        }
        if (tid == 0) { picks[k] = ri[0]; sv[ri[0]] = -1e30f; }
        __syncthreads();
    }
    if (tid < TOPKK && mask_gt[row] > 0.f) {
        int p = picks[tid];
        mask_pos[(size_t)row * LL + p] = mask_in[(size_t)row * LL + p];
    }
}

// ---------------------------------------------------------------------------
// K4: per (b,l): multi-GT resolution (argmax overlaps), fg + target_gt_idx.
// ---------------------------------------------------------------------------
__global__ void k_resolve(const float* __restrict__ overlaps,
                          float* __restrict__ mask_pos,
                          int* __restrict__ tgt,
                          float* __restrict__ fg) {
    const int idx = blockIdx.x * 256 + threadIdx.x;
    if (idx >= BB * LL) return;
    const int b = idx / LL, l = idx % LL;
    const size_t base = (size_t)b * NG * LL + l;

    float mp[NG]; float f = 0.f;
#pragma unroll
    for (int g = 0; g < NG; ++g) { mp[g] = mask_pos[base + (size_t)g * LL]; f += mp[g]; }

    int t = 0; float fgv;
    if (f > 1.0f) {
        float bo = -1e30f; int bg = 0;
#pragma unroll
        for (int g = 0; g < NG; ++g) {   // first-max semantics (strict >)
            float o = overlaps[base + (size_t)g * LL];
            if (o > bo) { bo = o; bg = g; }
        }
#pragma unroll
        for (int g = 0; g < NG; ++g) mask_pos[base + (size_t)g * LL] = (g == bg) ? 1.f : 0.f;
        t = bg; fgv = 1.0f;
    } else {
        fgv = f;
        float bm = mp[0];
#pragma unroll
        for (int g = 1; g < NG; ++g) if (mp[g] > bm) { bm = mp[g]; t = g; }
    }
    tgt[idx] = t;
    fg[idx]  = fgv;
}

// ---------------------------------------------------------------------------
// K5: per (b,g): pos_align = max_l align*mask_pos; pos_over = max_l ov*mask_pos
// ---------------------------------------------------------------------------
__global__ void k_rowmax(const float* __restrict__ align_raw,
                         const float* __restrict__ overlaps,
                         const float* __restrict__ mask_pos,
                         float* __restrict__ pos_align,
                         float* __restrict__ pos_over) {
    const int row = blockIdx.x;
    __shared__ float ra[256], ro[256];
    const int tid = threadIdx.x;
    const float* ar = align_raw + (size_t)row * LL;
    const float* ov = overlaps  + (size_t)row * LL;
    const float* mp = mask_pos  + (size_t)row * LL;
    float ma = 0.f, mo = 0.f;                 // all terms >= 0
    for (int i = tid; i < LL; i += 256) {
        float m = mp[i];
        ma = fmaxf(ma, ar[i] * m);
        mo = fmaxf(mo, ov[i] * m);
    }
    ra[tid] = ma; ro[tid] = mo;
    __syncthreads();
    for (int s = 128; s > 0; s >>= 1) {
        if (tid < s) { ra[tid] = fmaxf(ra[tid], ra[tid + s]); ro[tid] = fmaxf(ro[tid], ro[tid + s]); }
        __syncthreads();
    }
    if (tid == 0) { pos_align[row] = ra[0]; pos_over[row] = ro[0]; }
}

// ---------------------------------------------------------------------------
// K6: outputs, concatenated flat in return order:
// t_labels(BL) | t_bboxes(BL*5) | t_angles(BL) | t_scores(BL*80) | fg_mask(BL)
// ---------------------------------------------------------------------------
__global__ void k_finalize(const float* __restrict__ align_raw,
                           const float* __restrict__ mask_pos,
                           const float* __restrict__ pos_align,
                           const float* __restrict__ pos_over,
                           const int* __restrict__ gt_labels,
                           const float* __restrict__ gt_bboxes,
                           const float* __restrict__ gt_angles,
                           const int* __restrict__ tgt,
                           const float* __restrict__ fg,
                           float* __restrict__ out) {
    const int idx = blockIdx.x * 256 + threadIdx.x;
    if (idx >= BB * LL) return;
    const int b = idx / LL, l = idx % LL;
    const size_t pbase = (size_t)b * NG * LL + l;

    float norm = 0.f;
#pragma unroll
    for (int g = 0; g < NG; ++g) {
        int r = b * NG + g;
        float term = align_raw[pbase + (size_t)g * LL] * mask_pos[pbase + (size_t)g * LL]
                   * pos_over[r] / (pos_align[r] + 1e-9f);
        norm = fmaxf(norm, term);
    }
    const int t = tgt[idx];
    int lab = gt_labels[b * NG + t]; if (lab < 0) lab = 0;
    const float fgv = fg[idx];

    float* t_labels = out;
    float* t_bboxes = out + (size_t)BB * LL;
    float* t_angles = out + (size_t)BB * LL * 6;
    float* t_scores = out + (size_t)BB * LL * 7;
    float* fg_out   = out + (size_t)BB * LL * 87;

    t_labels[idx] = (float)lab;
    const float* gbb = gt_bboxes + ((size_t)b * NG + t) * 5;
#pragma unroll
    for (int c = 0; c < 5; ++c) t_bboxes[(size_t)idx * 5 + c] = gbb[c];
    t_angles[idx] = gt_angles[b * NG + t];

    const float val = (fgv > 0.f) ? norm : 0.f;
    float* srow = t_scores + (size_t)idx * NCLS;
#pragma unroll 8
    for (int c = 0; c < NCLS; ++c) srow[c] = (c == lab) ? val : 0.f;
    fg_out[idx] = (fgv > 0.f) ? 1.0f : 0.0f;
}

// ---------------------------------------------------------------------------
extern "C" void kernel_launch(void* const* d_in, const int* in_sizes, int n_in,
                              void* d_out, int out_size, void* d_ws, size_t ws_size,
                              hipStream_t stream) {
    const float* pd_scores = (const float*)d_in[0];   // (B,L,C)
    const float* pd_bboxes = (const float*)d_in[1];   // (B,L,5)
    /* d_in[2] pd_angles unused (== pd_bboxes[...,4]) */
    const float* anc       = (const float*)d_in[3];   // (1,L,2)
    const int*   gt_labels = (const int*)  d_in[4];   // (B,n,1)
    const float* gt_bboxes = (const float*)d_in[5];   // (B,n,5)
    const float* gt_angles = (const float*)d_in[6];   // (B,n,1)
    const float* mask_gt   = (const float*)d_in[7];   // (B,n,1)

    const size_t NE = (size_t)BB * NG * LL;  // 537600
    float* bbox_scores = (float*)d_ws;
    float* overlaps    = bbox_scores + NE;
    float* align_raw   = overlaps + NE;
    float* alignm      = align_raw + NE;
    float* mask_in     = alignm + NE;
    float* mask_pos    = mask_in + NE;
    float* pos_align   = mask_pos + NE;
    float* pos_over    = pos_align + BB * NG;
    float* fg          = pos_over + BB * NG;         // B*L
    int*   tgt         = (int*)(fg + (size_t)BB * LL);

    k_bbox_scores_wmma<<<BB * (LL / 16), 32, 0, stream>>>(pd_scores, gt_labels, bbox_scores);

    const int pairBlocks = (int)((NE + 255) / 256);  // 2100
    k_pairs<<<pairBlocks, 256, 0, stream>>>(pd_bboxes, anc, gt_bboxes, bbox_scores,
                                            overlaps, align_raw, alignm, mask_in, mask_pos);

    k_topk<<<BB * NG, 256, 0, stream>>>(alignm, mask_in, mask_gt, mask_pos);

    const int blBlocks = (BB * LL + 255) / 256;      // 132
    k_resolve<<<blBlocks, 256, 0, stream>>>(overlaps, mask_pos, tgt, fg);

    k_rowmax<<<BB * NG, 256, 0, stream>>>(align_raw, overlaps, mask_pos, pos_align, pos_over);

    k_finalize<<<blBlocks, 256, 0, stream>>>(align_raw, mask_pos, pos_align, pos_over,
                                             gt_labels, gt_bboxes, gt_angles, tgt, fg,
                                             (float*)d_out);
}